// TemporalGraphAttentionNetwork_86363202388376
// MI455X (gfx1250) — compile-verified
//
#include <hip/hip_runtime.h>
#include <hip/hip_bf16.h>

// ---------------- problem constants ----------------
static constexpr int NN   = 4096;    // nodes
static constexpr int EE   = 131072;  // edges
static constexpr int HID  = 256;
static constexpr int HH   = 4;       // heads
static constexpr int HD   = 64;      // head dim
static constexpr int IND  = 128;
static constexpr int OUTD = 128;

typedef __attribute__((ext_vector_type(2))) float v2f;
typedef __attribute__((ext_vector_type(8))) float v8f;

// D = A(16x4) * B(4x16) + C  -- fp32 WMMA, full precision
__device__ __forceinline__ v8f wmma4(v2f a, v2f b, v8f c) {
  return __builtin_amdgcn_wmma_f32_16x16x4_f32(false, a, false, b, (short)0, c,
                                               false, false);
}

// ---------------- tiled GEMM: C = A(MxK) * B(KxN) + bias ----------------
// compile-time K,N; one wave computes a 16x64 output strip (4 wmma tiles,
// A fragment reused 4x per k-step); blockDim = (32,8)
template <int K, int N>
__global__ void __launch_bounds__(256)
k_gemm(const float* __restrict__ A, const float* __restrict__ B,
       const float* __restrict__ bias, float* __restrict__ C, int M) {
  const int lane = threadIdx.x;
  const int half = lane >> 4;
  const int l15  = lane & 15;
  const int tn = blockIdx.x * 64;
  const int tm = (blockIdx.y * 8 + threadIdx.y) * 16;
  if (tm >= M) return;

  __builtin_prefetch(bias + tn + l15, 0, 1);

  v8f a0 = {}, a1 = {}, a2 = {}, a3 = {};
  const float* Arow = A + (size_t)(tm + l15) * K + (half ? 2 : 0);
  const float* Brow = B + (half ? 2 : 0) * (size_t)N + tn + l15;
#pragma unroll 8
  for (int kk = 0; kk < K; kk += 4) {
    const v2f a = *(const v2f*)Arow;
    Arow += 4;
    v2f b;
    b.x = Brow[0];  b.y = Brow[N];      a0 = wmma4(a, b, a0);
    b.x = Brow[16]; b.y = Brow[N + 16]; a1 = wmma4(a, b, a1);
    b.x = Brow[32]; b.y = Brow[N + 32]; a2 = wmma4(a, b, a2);
    b.x = Brow[48]; b.y = Brow[N + 48]; a3 = wmma4(a, b, a3);
    Brow += 4 * N;
  }
  const float b0 = bias[tn + l15];
  const float b1 = bias[tn + 16 + l15];
  const float b2 = bias[tn + 32 + l15];
  const float b3 = bias[tn + 48 + l15];
#pragma unroll
  for (int r = 0; r < 8; ++r) {
    float* crow = C + (size_t)(tm + r + half * 8) * N + tn + l15;
    crow[0]  = a0[r] + b0;
    crow[16] = a1[r] + b1;
    crow[32] = a2[r] + b2;
    crow[48] = a3[r] + b3;
  }
}

// ---------------- small helpers ----------------
__global__ void k_zero_f(float* p, int n) {
  int i = blockIdx.x * 256 + threadIdx.x;
  if (i < n) p[i] = 0.0f;
}
__global__ void k_zero_i(int* p, int n) {
  int i = blockIdx.x * 256 + threadIdx.x;
  if (i < n) p[i] = 0;
}

// h += t[n]*w_temp[c] + b_temp[c]
__global__ void k_add_temb(float* __restrict__ h, const float* __restrict__ t,
                           const float* __restrict__ wt, const float* __restrict__ bt) {
  int i = blockIdx.x * 256 + threadIdx.x;   // NN*HID total
  int n = i >> 8, c = i & 255;
  h[i] += t[n] * wt[c] + bt[c];
}

// ---------------- CSR build (dst-major) ----------------
__global__ void k_hist(const long long* __restrict__ ei, int* __restrict__ cnt) {
  int e = blockIdx.x * 256 + threadIdx.x;
  if (e < EE) atomicAdd(&cnt[(int)ei[EE + e]], 1);
}
__global__ void k_scan(const int* __restrict__ cnt, int* __restrict__ offs,
                       int* __restrict__ cursor) {
  __shared__ int part[256];
  const int t = threadIdx.x;
  const int base = t * 16;
  int s = 0;
  for (int j = 0; j < 16; ++j) s += cnt[base + j];
  part[t] = s;
  __syncthreads();
  for (int d = 1; d < 256; d <<= 1) {
    int v = (t >= d) ? part[t - d] : 0;
    __syncthreads();
    part[t] += v;
    __syncthreads();
  }
  int run = (t == 0) ? 0 : part[t - 1];
  for (int j = 0; j < 16; ++j) {
    int b = base + j;
    offs[b] = run;
    cursor[b] = run;
    run += cnt[b];
  }
  if (t == 255) offs[NN] = run;
}
__global__ void k_scatter(const long long* __restrict__ ei, int* __restrict__ cursor,
                          int* __restrict__ eid) {
  int e = blockIdx.x * 256 + threadIdx.x;
  if (e >= EE) return;
  int d = (int)ei[EE + e];
  int pos = atomicAdd(&cursor[d], 1);
  eid[pos] = e;
}

// ---------------- GAT pieces ----------------
// column sums of V
__global__ void k_colsum(const float* __restrict__ V, float* __restrict__ out) {
  __shared__ float red[256];
  const int col = blockIdx.x, t = threadIdx.x;
  float s = 0.0f;
  for (int r = t; r < NN; r += 256) s += V[(size_t)r * HID + col];
  red[t] = s;
  __syncthreads();
  for (int d = 128; d >= 1; d >>= 1) {
    if (t < d) red[t] += red[t + d];
    __syncthreads();
  }
  if (t == 0) out[col] = red[0];
}

// qa[n,h] = <Q[n,h,:], attn[h,:64]> ; ka[n,h] = <K[n,h,:], attn[h,64:]>
__global__ void k_alpha(const float* __restrict__ Q, const float* __restrict__ K,
                        const float* __restrict__ attn, float* __restrict__ qa,
                        float* __restrict__ ka) {
  __shared__ float rq[256], rk[256];
  const int n = blockIdx.x, t = threadIdx.x, hh = t >> 6, d = t & 63;
  rq[t] = Q[(size_t)n * HID + t] * attn[hh * 128 + d];
  rk[t] = K[(size_t)n * HID + t] * attn[hh * 128 + 64 + d];
  __syncthreads();
  for (int s = 32; s >= 1; s >>= 1) {
    if (d < s) { rq[t] += rq[t + s]; rk[t] += rk[t + s]; }
    __syncthreads();
  }
  if (d == 0) { qa[n * HH + hh] = rq[t]; ka[n * HH + hh] = rk[t]; }
}

// per-edge: w = exp(ew*leaky(qa+ka)) - 1 ; denom[dst,h] += w
__global__ void k_edge(const long long* __restrict__ ei, const float* __restrict__ ew,
                       const float* __restrict__ qa, const float* __restrict__ ka,
                       float* __restrict__ wbuf, float* __restrict__ denom) {
  int e = blockIdx.x * 256 + threadIdx.x;
  if (e >= EE) return;
  const int s = (int)ei[e];
  const int d = (int)ei[EE + e];
  const float w = ew[e];
#pragma unroll
  for (int hh = 0; hh < HH; ++hh) {
    float sc = qa[s * HH + hh] + ka[d * HH + hh];
    sc = sc >= 0.0f ? sc : 0.2f * sc;
    sc *= w;
    float v = __expf(sc) - 1.0f;
    wbuf[e * HH + hh] = v;
    atomicAdd(&denom[d * HH + hh], v);
  }
}

// per-dst aggregate + residual + LayerNorm(256)
__global__ void __launch_bounds__(256)
k_aggregate(const long long* __restrict__ ei, const int* __restrict__ offs,
            const int* __restrict__ eid, const float* __restrict__ wbuf,
            const float* __restrict__ V, const float* __restrict__ colsumV,
            const float* __restrict__ denom, const float* __restrict__ hin,
            const float* __restrict__ g, const float* __restrict__ b,
            float* __restrict__ hout) {
  __shared__ float red[256];
  const int n = blockIdx.x, t = threadIdx.x, hh = t >> 6;
  const float D = (float)NN + denom[n * HH + hh];
  float acc = colsumV[t];
  const int s0 = offs[n], s1 = offs[n + 1];
  for (int i = s0; i < s1; ++i) {
    int e = eid[i];
    int sn = (int)ei[e];
    acc += wbuf[e * HH + hh] * V[(size_t)sn * HID + t];
  }
  float x = acc / D + hin[(size_t)n * HID + t];
  red[t] = x;
  __syncthreads();
  for (int s = 128; s >= 1; s >>= 1) { if (t < s) red[t] += red[t + s]; __syncthreads(); }
  const float mean = red[0] * (1.0f / HID);
  __syncthreads();
  const float dx = x - mean;
  red[t] = dx * dx;
  __syncthreads();
  for (int s = 128; s >= 1; s >>= 1) { if (t < s) red[t] += red[t + s]; __syncthreads(); }
  const float var = red[0] * (1.0f / HID);
  hout[(size_t)n * HID + t] = dx * rsqrtf(var + 1e-5f) * g[t] + b[t];
}

// ---------------- flash self-attention (MHA pool) ----------------
// qkv: (NN, 768) = [q | k | v]; one wave per 16-query tile; 4 heads = 4 waves/block
__global__ void __launch_bounds__(128)
k_mha_flash(const float* __restrict__ qkv, float* __restrict__ O) {
  const int lane = threadIdx.x;
  const int half = lane >> 4, l15 = lane & 15;
  const int head = threadIdx.y;
  const int qb = blockIdx.x * 16;
  __shared__ float mS[HH][16], lS[HH][16], scS[HH][16];
  __shared__ float Ps[HH][16][17];
  if (lane < 16) { mS[head][lane] = -3.0e38f; lS[head][lane] = 0.0f; }
  __syncthreads();

  // hoist the loop-invariant Q fragment (pre-scaled by 1/sqrt(HD))
  v2f qr[16];
  {
    const float* qrow = qkv + (size_t)(qb + l15) * 768 + head * HD + (half ? 2 : 0);
#pragma unroll
    for (int kk = 0; kk < 16; ++kk) {
      v2f q = *(const v2f*)(qrow + 4 * kk);
      q.x *= 0.125f; q.y *= 0.125f;
      qr[kk] = q;
    }
  }

  v8f o0 = {}, o1 = {}, o2 = {}, o3 = {};

  for (int kb = 0; kb < NN; kb += 16) {
    // S = (q*scale) @ K^T   (16x64 @ 64x16)
    v8f s = {};
    const float* krow = qkv + (size_t)(kb + l15) * 768 + HID + head * HD + (half ? 2 : 0);
#pragma unroll
    for (int kk = 0; kk < 16; ++kk) {
      const v2f b = *(const v2f*)(krow + 4 * kk);
      s = wmma4(qr[kk], b, s);
    }
    // online softmax; lane holds rows r+half*8, column l15
#pragma unroll
    for (int r = 0; r < 8; ++r) {
      const int row = r + half * 8;
      float v = s[r];
      for (int m = 8; m >= 1; m >>= 1) v = fmaxf(v, __shfl_xor(v, m, 16));
      const float mold = mS[head][row];
      const float mnew = fmaxf(mold, v);
      const float p = __expf(s[r] - mnew);
      float ps = p;
      for (int m = 8; m >= 1; m >>= 1) ps += __shfl_xor(ps, m, 16);
      if (l15 == 0) {
        const float sc = __expf(mold - mnew);
        scS[head][row] = sc;
        mS[head][row] = mnew;
        lS[head][row] = lS[head][row] * sc + ps;
      }
      Ps[head][row][l15] = p;
    }
    __syncthreads();
    // rescale accumulators, then O += P(16x16) @ Vtile(16x64)
#pragma unroll
    for (int r = 0; r < 8; ++r) {
      const float sc = scS[head][r + half * 8];
      o0[r] *= sc; o1[r] *= sc; o2[r] *= sc; o3[r] *= sc;
    }
#pragma unroll
    for (int kk = 0; kk < 16; kk += 4) {
      const int off = kk + (half ? 2 : 0);
      v2f a;
      a.x = Ps[head][l15][off];
      a.y = Ps[head][l15][off + 1];
      const float* vb = qkv + (size_t)(kb + off) * 768 + 2 * HID + head * HD + l15;
      v2f b;
      b.x = vb[0];  b.y = vb[768];      o0 = wmma4(a, b, o0);
      b.x = vb[16]; b.y = vb[768 + 16]; o1 = wmma4(a, b, o1);
      b.x = vb[32]; b.y = vb[768 + 32]; o2 = wmma4(a, b, o2);
      b.x = vb[48]; b.y = vb[768 + 48]; o3 = wmma4(a, b, o3);
    }
    __syncthreads();
  }
#pragma unroll
  for (int r = 0; r < 8; ++r) {
    const int row = r + half * 8;
    const float inv = 1.0f / lS[head][row];
    float* orow = O + (size_t)(qb + row) * HID + head * HD;
    orow[l15]      = o0[r] * inv;
    orow[16 + l15] = o1[r] * inv;
    orow[32 + l15] = o2[r] * inv;
    orow[48 + l15] = o3[r] * inv;
  }
}

// final LayerNorm over 128
__global__ void k_ln128(const float* __restrict__ X, const float* __restrict__ g,
                        const float* __restrict__ b, float* __restrict__ out) {
  __shared__ float red[128];
  const int n = blockIdx.x, t = threadIdx.x;
  const float x = X[(size_t)n * OUTD + t];
  red[t] = x;
  __syncthreads();
  for (int s = 64; s >= 1; s >>= 1) { if (t < s) red[t] += red[t + s]; __syncthreads(); }
  const float mean = red[0] * (1.0f / OUTD);
  __syncthreads();
  const float dx = x - mean;
  red[t] = dx * dx;
  __syncthreads();
  for (int s = 64; s >= 1; s >>= 1) { if (t < s) red[t] += red[t + s]; __syncthreads(); }
  const float var = red[0] * (1.0f / OUTD);
  out[(size_t)n * OUTD + t] = dx * rsqrtf(var + 1e-5f) * g[t] + b[t];
}

// ---------------- host orchestration ----------------
extern "C" void kernel_launch(void* const* d_in, const int* in_sizes, int n_in,
                              void* d_out, int out_size, void* d_ws, size_t ws_size,
                              hipStream_t stream) {
  const float*     x    = (const float*)d_in[0];
  const long long* ei   = (const long long*)d_in[1];   // int64 (2,E)
  const float*     ew   = (const float*)d_in[2];
  const float*     tpos = (const float*)d_in[3];
  const float*     inW  = (const float*)d_in[4];
  const float*     inB  = (const float*)d_in[5];
  const float*     tW   = (const float*)d_in[6];
  const float*     tB   = (const float*)d_in[7];
  // layers: 8 + l*9 : wq,bq,wk,bk,wv,bv,attn,ln_g,ln_b
  const float* mhaW  = (const float*)d_in[26];
  const float* mhaB  = (const float*)d_in[27];
  const float* outW  = (const float*)d_in[28];
  const float* outB  = (const float*)d_in[29];
  const float* projW = (const float*)d_in[30];
  const float* projB = (const float*)d_in[31];
  const float* lnG   = (const float*)d_in[32];
  const float* lnB   = (const float*)d_in[33];

  float* ws = (float*)d_ws;
  float* h      = ws;                       // NN*HID
  float* h2     = ws + 1048576;             // NN*HID
  float* big    = ws + 2097152;             // 3*NN*HID (Q|K|V -> qkv -> pooled/lin)
  float* Qb = big, *Kb = big + 1048576, *Vb = big + 2097152;
  float* wbuf   = ws + 5242880;             // EE*HH
  float* denom  = ws + 5767168;             // NN*HH
  float* colsum = ws + 5783552;             // HID
  float* qa     = ws + 5783808;             // NN*HH
  float* ka     = ws + 5800192;             // NN*HH
  int*   iw     = (int*)(ws + 5816576);
  int* cnt = iw, *offs = iw + NN, *cursor = iw + NN + NN + 1, *eid = cursor + NN;

  const dim3 tb(32, 8);

  // CSR (edges identical across layers: build once per launch)
  k_zero_i<<<16, 256, 0, stream>>>(cnt, NN);
  k_hist<<<EE / 256, 256, 0, stream>>>(ei, cnt);
  k_scan<<<1, 256, 0, stream>>>(cnt, offs, cursor);
  k_scatter<<<EE / 256, 256, 0, stream>>>(ei, cursor, eid);

  // h = x @ W_in + b ; += temporal embedding
  k_gemm<IND, HID><<<dim3(HID / 64, NN / 128), tb, 0, stream>>>(x, inW, inB, h, NN);
  k_add_temb<<<NN, 256, 0, stream>>>(h, tpos, tW, tB);

  float* cur = h;
  float* alt = h2;
  for (int l = 0; l < 2; ++l) {
    const float* wq  = (const float*)d_in[8 + l * 9 + 0];
    const float* bq  = (const float*)d_in[8 + l * 9 + 1];
    const float* wk  = (const float*)d_in[8 + l * 9 + 2];
    const float* bk  = (const float*)d_in[8 + l * 9 + 3];
    const float* wv  = (const float*)d_in[8 + l * 9 + 4];
    const float* bv  = (const float*)d_in[8 + l * 9 + 5];
    const float* att = (const float*)d_in[8 + l * 9 + 6];
    const float* lg  = (const float*)d_in[8 + l * 9 + 7];
    const float* lb  = (const float*)d_in[8 + l * 9 + 8];

    k_gemm<HID, HID><<<dim3(HID / 64, NN / 128), tb, 0, stream>>>(cur, wq, bq, Qb, NN);
    k_gemm<HID, HID><<<dim3(HID / 64, NN / 128), tb, 0, stream>>>(cur, wk, bk, Kb, NN);
    k_gemm<HID, HID><<<dim3(HID / 64, NN / 128), tb, 0, stream>>>(cur, wv, bv, Vb, NN);
    k_colsum<<<HID, 256, 0, stream>>>(Vb, colsum);
    k_alpha<<<NN, 256, 0, stream>>>(Qb, Kb, att, qa, ka);
    k_zero_f<<<(NN * HH) / 256, 256, 0, stream>>>(denom, NN * HH);
    k_edge<<<EE / 256, 256, 0, stream>>>(ei, ew, qa, ka, wbuf, denom);
    k_aggregate<<<NN, 256, 0, stream>>>(ei, offs, eid, wbuf, Vb, colsum, denom, cur,
                                        lg, lb, alt);
    float* tmp = cur; cur = alt; alt = tmp;
  }

  // MHA pool: qkv = cur @ in_w + in_b  (N x 768)
  k_gemm<HID, 3 * HID><<<dim3(768 / 64, NN / 128), tb, 0, stream>>>(cur, mhaW, mhaB,
                                                                    big, NN);
  k_mha_flash<<<dim3(NN / 16, 1), dim3(32, 4), 0, stream>>>(big, alt);
  // pooled = o @ out_w + out_b
  k_gemm<HID, HID><<<dim3(HID / 64, NN / 128), tb, 0, stream>>>(alt, outW, outB, big, NN);
  // lin = pooled @ out_proj + b
  k_gemm<HID, OUTD><<<dim3(OUTD / 64, NN / 128), tb, 0, stream>>>(big, projW, projB,
                                                                  big + 1048576, NN);
  k_ln128<<<NN, 128, 0, stream>>>(big + 1048576, lnG, lnB, (float*)d_out);
}